// ROIRotate_78434692760170
// MI455X (gfx1250) — compile-verified
//
#include <hip/hip_runtime.h>
#include <cstddef>

typedef __attribute__((ext_vector_type(2))) float v2f;
typedef __attribute__((ext_vector_type(8))) float v8f;

#define CROP_H 28
#define OUT_HW 224
#define FH 256
#define FW 256
#define MAXN 64
#define HALF_M 112                 // half of the 224 output rows per block
#define LDSTRIDE 260               // FW + 4 pad -> avoid 64-bank LDS conflicts
#define LDS_BYTES (HALF_M * LDSTRIDE * 4)

struct RoiParams {
    float ax[MAXN], cx[MAXN], ay[MAXN], cy[MAXN];
    int   wb[MAXN];
    int   order[MAXN];
    int   maxw;
};

// ---------------------------------------------------------------------------
// Kernel 1: per-box affine params, widths, maxw, and descending-width order.
// ---------------------------------------------------------------------------
__global__ void roi_setup_kernel(const int* __restrict__ boxes, RoiParams* P, int N) {
    if (threadIdx.x != 0 || blockIdx.x != 0) return;
    int maxw = 0;
    for (int n = 0; n < N; ++n) {
        int x1 = boxes[n * 4 + 0], y1 = boxes[n * 4 + 1];
        int x3 = boxes[n * 4 + 2], y3 = boxes[n * 4 + 3];
        int bw = x3 - x1; if (bw < 0) bw = -bw;
        int bh = y3 - y1; if (bh < 0) bh = -bh;
        int boxw = bw > bh ? bw : bh;
        int boxh = bw > bh ? bh : bw;
        int wb = (CROP_H * boxw + boxh - 1) / boxh;   // ceil(28*box_w/box_h)
        if (wb > FW) wb = FW;
        P->wb[n] = wb;
        P->ax[n] = (float)(x3 - x1) / (float)wb;
        P->cx[n] = (float)x1 * ((float)(FW - 1) / (float)FW);
        P->ay[n] = (float)(y3 - y1) / (float)CROP_H;
        P->cy[n] = (float)y1 * ((float)(FH - 1) / (float)FH);
        if (wb > maxw) maxw = wb;
    }
    P->maxw = maxw;
    // order = np.argsort(widths)[::-1]  (stable ascending, reversed):
    // descending widths; among ties the LARGER index comes first.
    bool used[MAXN];
    for (int i = 0; i < N; ++i) used[i] = false;
    for (int r = 0; r < N; ++r) {
        int best = -1, bestw = -1;
        for (int i = 0; i < N; ++i)
            if (!used[i] && P->wb[i] >= bestw) { bestw = P->wb[i]; best = i; }
        used[best] = true;
        P->order[r] = best;
    }
}

// ---------------------------------------------------------------------------
// Kernel 2: build dense row operators.
//   Ay [N][224][256]  : Ay[n][p][Y] = sum_h My[p,h] * Wy(n,h,Y)
//   AxT[N][256][224]  : AxT[n][X][q] = sum_w Mx[q,w] * mask(n,w) * Wx(n,w,X)
// ---------------------------------------------------------------------------
__global__ void roi_build_kernel(const RoiParams* __restrict__ P,
                                 float* __restrict__ Ay, float* __restrict__ AxT, int N) {
    int tid = blockIdx.x * blockDim.x + threadIdx.x;
    int half = N * OUT_HW;
    if (tid < half) {
        int n = tid / OUT_HW, p = tid % OUT_HW;
        float ay = P->ay[n], cy = P->cy[n];
        float* row = Ay + (size_t)(n * OUT_HW + p) * FW;
        for (int Y = 0; Y < FH; ++Y) row[Y] = 0.f;
        float pos = (float)(CROP_H - 1) * (float)p / (float)(OUT_HW - 1);
        int i0 = (int)floorf(pos);
        if (i0 > CROP_H - 2) i0 = CROP_H - 2;
        if (i0 < 0) i0 = 0;
        float t = pos - (float)i0;
        for (int j = 0; j < 2; ++j) {
            int h = i0 + j;
            float m = j ? t : (1.f - t);
            float iy = ay * (float)h + cy;
            float yf = floorf(iy);
            int y0 = (int)yf;
            float wy = iy - yf;
            if (y0 >= 0 && y0 < FH)         row[y0]     += (1.f - wy) * m;
            if (y0 + 1 >= 0 && y0 + 1 < FH) row[y0 + 1] += wy * m;
        }
    } else if (tid < 2 * half) {
        int t2 = tid - half;
        int n = t2 / OUT_HW, q = t2 % OUT_HW;
        float ax = P->ax[n], cx = P->cx[n];
        int wb = P->wb[n], maxw = P->maxw;
        float* col = AxT + (size_t)n * FW * OUT_HW + q;
        for (int X = 0; X < FW; ++X) col[(size_t)X * OUT_HW] = 0.f;
        float pos = (float)(maxw - 1) * (float)q / (float)(OUT_HW - 1);
        int i0 = (int)floorf(pos);
        if (i0 > maxw - 2) i0 = maxw - 2;
        if (i0 < 0) i0 = 0;
        float t = pos - (float)i0;
        for (int j = 0; j < 2; ++j) {
            int w = i0 + j;
            float m = j ? t : (1.f - t);
            if (w < wb) {
                float ix = ax * (float)w + cx;
                float xf = floorf(ix);
                int x0 = (int)xf;
                float wx = ix - xf;
                if (x0 >= 0 && x0 < FW)         col[(size_t)x0 * OUT_HW]       += (1.f - wx) * m;
                if (x0 + 1 >= 0 && x0 + 1 < FW) col[(size_t)(x0 + 1) * OUT_HW] += wx * m;
            }
        }
    }
}

// ---------------------------------------------------------------------------
// Kernel 3: out[n',c] = Ay[n] @ feat[c] @ AxT[n]   via V_WMMA_F32_16X16X4_F32.
// One block = (n', c, half-of-p).  T = Ay_half @ feat staged in LDS.
// Tile loops are driven by a readfirstlane'd wave index so the compiler emits
// scalar (uniform) loops -> EXEC is structurally all-1s around every WMMA.
// ---------------------------------------------------------------------------
__global__ __launch_bounds__(256) void roi_gemm_kernel(
        const float* __restrict__ feat,
        const float* __restrict__ Ay,
        const float* __restrict__ AxT,
        const RoiParams* __restrict__ P,
        float* __restrict__ out, int C) {
    extern __shared__ float Tsh[];          // [HALF_M][LDSTRIDE]

    int b = blockIdx.x;
    int halfIdx = b & 1;
    int rest = b >> 1;
    int c  = rest % C;
    int np = rest / C;
    int n  = P->order[np];
    int m_off = halfIdx * HALF_M;

    // wave index is identical across the 32 lanes of a wave; make it scalar.
    int wave = __builtin_amdgcn_readfirstlane((int)(threadIdx.x >> 5));
    int lane = threadIdx.x & 31;
    int l  = lane & 15;
    int hs = lane >> 4;                      // 0: lanes 0-15, 1: lanes 16-31

    const float* AyN = Ay + (size_t)(n * OUT_HW + m_off) * FW;
    const float* fc  = feat + (size_t)c * FH * FW;
    const float* AxN = AxT + (size_t)n * FW * OUT_HW;
    float* outBase   = out + ((size_t)(np * C + c) * OUT_HW + m_off) * OUT_HW;

    // ---- GEMM1: T[112][256] = Ay_half[112][256] @ feat[256][256] ----
    const int TILES1 = (HALF_M / 16) * (FW / 16);     // 7 * 16 = 112
    for (int t = wave; t < TILES1; t += 8) {          // scalar uniform loop
        int m0 = (t / 16) * 16;
        int n0 = (t % 16) * 16;
        v8f acc = {};
        const float* arow = AyN + (size_t)(m0 + l) * FW;
        for (int k = 0; k < FH; k += 4) {
            int ka = k + 2 * hs;
            // A 16x4 f32 frag: V0 = K0(lo lanes)/K2(hi lanes), V1 = K1/K3
            v2f a; a.x = arow[ka]; a.y = arow[ka + 1];
            // B 4x16 f32 frag: rows striped across lanes; hi lanes = K+2
            v2f bv; bv.x = fc[(size_t)ka * FW + n0 + l];
                    bv.y = fc[(size_t)(ka + 1) * FW + n0 + l];
            acc = __builtin_amdgcn_wmma_f32_16x16x4_f32(
                      false, a, false, bv, (short)0, acc, false, false);
        }
        // C/D 16x16 f32: VGPR r -> row r (lo lanes) / row r+8 (hi lanes)
        #pragma unroll
        for (int r = 0; r < 8; ++r)
            Tsh[(m0 + r + 8 * hs) * LDSTRIDE + n0 + l] = acc[r];
    }
    __syncthreads();

    // ---- GEMM2: out_half[112][224] = T[112][256] @ AxT[n][256][224] ----
    const int TILES2 = (HALF_M / 16) * (OUT_HW / 16); // 7 * 14 = 98
    for (int t = wave; t < TILES2; t += 8) {
        int m0 = (t / 14) * 16;
        int q0 = (t % 14) * 16;
        v8f acc = {};
        const float* arow = Tsh + (size_t)(m0 + l) * LDSTRIDE;
        for (int k = 0; k < FW; k += 4) {
            int ka = k + 2 * hs;
            v2f a; a.x = arow[ka]; a.y = arow[ka + 1];
            v2f bv; bv.x = AxN[(size_t)ka * OUT_HW + q0 + l];
                    bv.y = AxN[(size_t)(ka + 1) * OUT_HW + q0 + l];
            acc = __builtin_amdgcn_wmma_f32_16x16x4_f32(
                      false, a, false, bv, (short)0, acc, false, false);
        }
        #pragma unroll
        for (int r = 0; r < 8; ++r)
            outBase[(size_t)(m0 + r + 8 * hs) * OUT_HW + q0 + l] = acc[r];
    }
}

// ---------------------------------------------------------------------------
extern "C" void kernel_launch(void* const* d_in, const int* in_sizes, int n_in,
                              void* d_out, int out_size, void* d_ws, size_t ws_size,
                              hipStream_t stream) {
    const float* feature = (const float*)d_in[0];
    const int*   boxes   = (const int*)d_in[1];
    (void)n_in; (void)out_size; (void)ws_size;

    int C = in_sizes[0] / (FH * FW);   // 32
    int N = in_sizes[1] / 4;           // 32

    RoiParams* P = (RoiParams*)d_ws;
    float* Ay  = (float*)((char*)d_ws + 8192);
    float* AxT = Ay + (size_t)N * OUT_HW * FW;
    float* outp = (float*)d_out;

    hipFuncSetAttribute((const void*)roi_gemm_kernel,
                        hipFuncAttributeMaxDynamicSharedMemorySize, LDS_BYTES);

    roi_setup_kernel<<<1, 32, 0, stream>>>(boxes, P, N);

    int buildThreads = 2 * N * OUT_HW;
    roi_build_kernel<<<(buildThreads + 255) / 256, 256, 0, stream>>>(P, Ay, AxT, N);

    roi_gemm_kernel<<<N * C * 2, 256, LDS_BYTES, stream>>>(feature, Ay, AxT, P, outp, C);
}